// PerTokenAttentionProbe_35433480192225
// MI455X (gfx1250) — compile-verified
//
#include <hip/hip_runtime.h>
#include <hip/hip_bf16.h>
#include <math.h>

// Problem constants (match reference)
#define HID 2048
#define SEQ 2048
#define NB  4
#define NH  4
#define NCOL 16   // padded output columns: [0..3]=q_logits, [4..7]=values, [8..15]=0

#define NCHUNK (HID / 8)          // 8-wide K-chunks = 256
#define QV_FLOATS   (NB * SEQ * NCOL)      // 131072
#define BFRAG_FLOATS (NCHUNK * 32 * 4)     // 32768

typedef float v2f __attribute__((ext_vector_type(2)));
typedef float v8f __attribute__((ext_vector_type(8)));

// Combined virtual weight matrix W[H][16] = [Wq | Wv | zeros]
__device__ __forceinline__ float weight_col(const float* __restrict__ Wq,
                                            const float* __restrict__ Wv,
                                            int k, int n) {
    if (n < 4)      return Wq[k * 4 + n];
    else if (n < 8) return Wv[k * 4 + (n - 4)];
    else            return 0.0f;
}

// ---------------------------------------------------------------------------
// Prep: materialize pre-permuted B fragments + padded bias (one-time, tiny).
// bfrag[chunk][lane][0..3]: for chunk kc/8, lane (kk,n), the four W values
//   W[kc + kk*4 + 0..3][n]  (same K permutation the GEMM applies to A).
// ---------------------------------------------------------------------------
__global__ __launch_bounds__(256) void prep_bfrag(
    const float* __restrict__ Wq, const float* __restrict__ bq,
    const float* __restrict__ Wv, const float* __restrict__ bv,
    float* __restrict__ bfrag,    // [NCHUNK*32*4]
    float* __restrict__ biaspad)  // [16]
{
    const int t = blockIdx.x * 256 + threadIdx.x;   // 0..NCHUNK*32-1
    const int c = t >> 5;
    const int l = t & 31;
    const int kk = l >> 4;
    const int n  = l & 15;
    const int kbase = c * 8 + kk * 4;

    float4 w;
    w.x = weight_col(Wq, Wv, kbase + 0, n);
    w.y = weight_col(Wq, Wv, kbase + 1, n);
    w.z = weight_col(Wq, Wv, kbase + 2, n);
    w.w = weight_col(Wq, Wv, kbase + 3, n);
    *reinterpret_cast<float4*>(bfrag + (size_t)t * 4) = w;

    if (t < 16) biaspad[t] = (t < 4) ? bq[t] : ((t < 8) ? bv[t - 4] : 0.0f);
}

// ---------------------------------------------------------------------------
// Pass 1: qv[row][0..15] = hidden[row,:] @ [Wq|Wv|0] + [bq|bv|0]
// One wave per 16-row tile.  Inner loop: 16 K per iteration =
// 2 coalesced A b128 loads (HBM) + 2 B-fragment b128 loads (L2-hot) +
// 4 x V_WMMA_F32_16X16X4_F32.  Branch-free body with speculative prefetch.
// ---------------------------------------------------------------------------
__global__ __launch_bounds__(256) void qv_gemm_wmma(
    const float* __restrict__ hidden,   // [NB*SEQ, HID]
    const float* __restrict__ bfrag,    // [NCHUNK*32*4]
    const float* __restrict__ biaspad,  // [16]
    float* __restrict__ qv)             // [NB*SEQ, NCOL]
{
    const int lane = threadIdx.x & 31;
    const int wave = threadIdx.x >> 5;
    const int tile = blockIdx.x * 8 + wave;   // 8 waves per block
    const int row0 = tile * 16;               // 16 rows per tile

    const int m  = lane & 15;   // A-matrix row within tile
    const int kk = lane >> 4;   // lane half: K slots {0,1} vs {2,3}
    const int n  = lane & 15;   // B/D column

    const float* arow  = hidden + (size_t)(row0 + m) * HID + kk * 4;
    const float* bptr  = bfrag + (size_t)lane * 4;

    v8f acc = {};

    #pragma unroll 4
    for (int c = 0; c < HID / 16; ++c) {      // 128 iterations, 16 K each
        const float4 a0 = *reinterpret_cast<const float4*>(arow + c * 16);
        const float4 a1 = *reinterpret_cast<const float4*>(arow + c * 16 + 8);
        const float4 b0 = *reinterpret_cast<const float4*>(bptr + (size_t)(2 * c + 0) * 128);
        const float4 b1 = *reinterpret_cast<const float4*>(bptr + (size_t)(2 * c + 1) * 128);

        // speculative prefetch ~1KB ahead (harmlessly dropped past row end)
        __builtin_prefetch(arow + c * 16 + 256, 0, 0);

        v2f aA0 = { a0.x, a0.y }, aB0 = { a0.z, a0.w };
        v2f bA0 = { b0.x, b0.y }, bB0 = { b0.z, b0.w };
        v2f aA1 = { a1.x, a1.y }, aB1 = { a1.z, a1.w };
        v2f bA1 = { b1.x, b1.y }, bB1 = { b1.z, b1.w };

        acc = __builtin_amdgcn_wmma_f32_16x16x4_f32(false, aA0, false, bA0,
                                                    (short)0, acc, false, false);
        acc = __builtin_amdgcn_wmma_f32_16x16x4_f32(false, aB0, false, bB0,
                                                    (short)0, acc, false, false);
        acc = __builtin_amdgcn_wmma_f32_16x16x4_f32(false, aA1, false, bA1,
                                                    (short)0, acc, false, false);
        acc = __builtin_amdgcn_wmma_f32_16x16x4_f32(false, aB1, false, bB1,
                                                    (short)0, acc, false, false);
    }

    const float bias = biaspad[n];

    // D layout: VGPR r -> M = r (lanes 0-15) or M = 8+r (lanes 16-31), N = n
    #pragma unroll
    for (int r = 0; r < 8; ++r) {
        const int row = row0 + kk * 8 + r;
        qv[(size_t)row * NCOL + n] = acc[r] + bias;
    }
}

// ---------------------------------------------------------------------------
// Pass 2: out[b,i] = (1/NH) * sum_h sum_j softmax_j(w_h*relu(j-i)) * v[b,j,h]
// q_logits cancel inside softmax (shift invariance); softmax max is analytic:
// M = max(0, w_h*(S-1-i)).  One wave per query i; all 4 batches accumulated
// simultaneously (attention weights are batch-independent); values are
// L2-resident (512 KB).
// ---------------------------------------------------------------------------
__global__ __launch_bounds__(256) void attn_reduce(
    const float* __restrict__ qv,     // [NB*SEQ, NCOL]; values at cols 4..7
    const float* __restrict__ pos_w,  // [NH]
    float* __restrict__ out)          // [NB*SEQ]  (= [B,S,1])
{
    const int lane = threadIdx.x & 31;
    const int wave = threadIdx.x >> 5;
    const int i    = blockIdx.x * 8 + wave;   // query index

    float outacc[NB] = {0.f, 0.f, 0.f, 0.f};

    #pragma unroll
    for (int h = 0; h < NH; ++h) {
        const float w = pos_w[h];
        float M = w * (float)(SEQ - 1 - i);
        if (M < 0.f) M = 0.f;

        float l = 0.f;
        float acc[NB] = {0.f, 0.f, 0.f, 0.f};

        for (int j = lane; j < SEQ; j += 32) {
            const int d = j - i;
            const float pos = (d > 0) ? (float)d : 0.0f;
            const float p = __expf(w * pos - M);   // argument <= 0, stable
            l += p;
            #pragma unroll
            for (int b = 0; b < NB; ++b) {
                acc[b] += p * qv[(size_t)(b * SEQ + j) * NCOL + 4 + h];
            }
        }

        // wave32 reduction
        #pragma unroll
        for (int off = 16; off > 0; off >>= 1) {
            l += __shfl_xor(l, off, 32);
            #pragma unroll
            for (int b = 0; b < NB; ++b)
                acc[b] += __shfl_xor(acc[b], off, 32);
        }

        const float inv = 1.0f / l;
        #pragma unroll
        for (int b = 0; b < NB; ++b)
            outacc[b] += acc[b] * inv;
    }

    if (lane == 0) {
        #pragma unroll
        for (int b = 0; b < NB; ++b)
            out[b * SEQ + i] = 0.25f * outacc[b];
    }
}

extern "C" void kernel_launch(void* const* d_in, const int* in_sizes, int n_in,
                              void* d_out, int out_size, void* d_ws, size_t ws_size,
                              hipStream_t stream) {
    const float* hidden = (const float*)d_in[0];
    const float* Wq     = (const float*)d_in[1];
    const float* bq     = (const float*)d_in[2];
    const float* Wv     = (const float*)d_in[3];
    const float* bv     = (const float*)d_in[4];
    const float* pos_w  = (const float*)d_in[5];
    float* out = (float*)d_out;

    float* qv      = (float*)d_ws;            // 512 KB
    float* bfrag   = qv + QV_FLOATS;          // 128 KB
    float* biaspad = bfrag + BFRAG_FLOATS;    // 64 B

    // Prep: NCHUNK*32 = 8192 threads -> 32 blocks
    prep_bfrag<<<32, 256, 0, stream>>>(Wq, bq, Wv, bv, bfrag, biaspad);
    // Pass 1: 8192 rows / (8 waves * 16 rows) = 64 blocks
    qv_gemm_wmma<<<64, 256, 0, stream>>>(hidden, bfrag, biaspad, qv);
    // Pass 2: 2048 queries / 8 waves = 256 blocks
    attn_reduce<<<256, 256, 0, stream>>>(qv, pos_w, out);
}